// TemporalFactorizedBlock_43937515438135
// MI455X (gfx1250) — compile-verified
//
#include <hip/hip_runtime.h>
#include <math.h>

typedef __attribute__((ext_vector_type(2))) float v2f;
typedef __attribute__((ext_vector_type(8))) float v8f;

#define WMMA_F32(a, b, c) \
    __builtin_amdgcn_wmma_f32_16x16x4_f32(false, (a), false, (b), (short)0, (c), false, false)

#if defined(__gfx1250__) && __has_builtin(__builtin_amdgcn_global_load_async_to_lds_b128) && \
    __has_builtin(__builtin_amdgcn_s_wait_asynccnt)
#define HAS_ASYNC_LDS 1
#else
#define HAS_ASYNC_LDS 0
#endif

typedef __attribute__((ext_vector_type(4))) int v4i;
typedef v4i __attribute__((address_space(1)))* gv4i_p;   // ptr to int4 in global AS
typedef v4i __attribute__((address_space(3)))* lv4i_p;   // ptr to int4 in LDS AS

// 16-byte global -> LDS copy; async (ASYNCcnt-tracked) when the builtin exists.
__device__ __forceinline__ void copy16_g2l(const float* gp, float* lp)
{
#if HAS_ASYNC_LDS
    __builtin_amdgcn_global_load_async_to_lds_b128((gv4i_p)gp, (lv4i_p)lp, 0, 0);
#else
    *(float4*)lp = *(const float4*)gp;
#endif
}

__device__ __forceinline__ void wait_async_copies()
{
#if HAS_ASYNC_LDS
    __builtin_amdgcn_s_wait_asynccnt(0);
#endif
}

constexpr int BATCH = 4;
constexpr int SEQ   = 2049;
constexpr int DIM   = 256;
constexpr int NH    = 4;
constexpr int DH    = 64;          // DIM / NH
constexpr int WIN   = 64;
constexpr int QKV   = 3 * DIM;     // 768
constexpr int FF    = 512;         // 2 * DIM
constexpr int ROWS  = BATCH * SEQ; // 8196
constexpr int QT    = (SEQ + 15) / 16;  // 129

// ---------------------------------------------------------------------------
// Generic fused GEMM:  out[m, n] = act( rms?(A)[m,:] @ W[n,:] + bias[n] ) + resid
// A: [ROWS, KD] row-major, W: [N, KD] row-major (torch Linear weight -> A@W^T).
// 256 threads = 8 waves; MB rows per block; each wave owns 16 cols and MB/16
// accumulators so one B fragment feeds MB/16 v_wmma_f32_16x16x4_f32 ops.
// A tile staged in LDS via ASYNCcnt-tracked global_load_async_to_lds_b128.
// ---------------------------------------------------------------------------
template <int KD, int MB, bool RMS, bool GELU, bool RESID>
__global__ void __launch_bounds__(256)
k_gemm(const float* __restrict__ A, const float* __restrict__ nw,
       const float* __restrict__ W, const float* __restrict__ bias,
       const float* __restrict__ resid, float* __restrict__ out, int nstride)
{
    constexpr int AS  = KD + 4;        // LDS row stride in floats (16B-aligned pad)
    constexpr int SUB = MB / 16;       // 16-row subtiles per block
    constexpr int TPR = 256 / MB;      // staging threads per row
    constexpr int CH  = KD / TPR;      // floats staged per thread (== 32)
    __shared__ float At[MB * AS];
    __shared__ float red[256];
    __shared__ float rs[MB];

    const int tid  = threadIdx.x;
    const int lane = tid & 31;
    const int wave = tid >> 5;
    const int half = lane >> 4;
    const int ln   = lane & 15;
    const int m0   = blockIdx.x * MB;

    // ---- stage MB x KD tile of A into LDS (async DMA path when available) ----
    {
        const int r   = tid / TPR;
        const int c0  = (tid % TPR) * CH;
        const int row = min(m0 + r, ROWS - 1);
        const float* ap = A + (size_t)row * KD + c0;
        float* lp = At + r * AS + c0;
        #pragma unroll
        for (int j = 0; j < CH; j += 4)
            copy16_g2l(ap + j, lp + j);
        wait_async_copies();
        __syncthreads();
        if (RMS) {
            float ss = 0.f;
            #pragma unroll
            for (int j = 0; j < CH; ++j) { const float v = lp[j]; ss += v * v; }
            red[tid] = ss;
            __syncthreads();
            if (tid < MB) {
                float s = 0.f;
                #pragma unroll
                for (int j = 0; j < TPR; ++j) s += red[tid * TPR + j];
                rs[tid] = rsqrtf(s / (float)KD + 1e-6f);
            }
            __syncthreads();
            const float sc = rs[r];
            #pragma unroll
            for (int j = 0; j < CH; ++j)
                lp[j] = lp[j] * sc * nw[c0 + j];
            __syncthreads();
        }
    }

    // ---- WMMA K-loop: 1 B fragment -> SUB wmma ops ----
    const int n = blockIdx.y * 128 + wave * 16 + ln;
    const float2* wp = (const float2*)(W + (size_t)n * KD);

    v8f acc[SUB];
    #pragma unroll
    for (int s = 0; s < SUB; ++s)
        #pragma unroll
        for (int r = 0; r < 8; ++r) acc[s][r] = 0.f;

    #pragma unroll 8
    for (int k0 = 0; k0 < KD; k0 += 4) {
        const float2 bw = wp[(k0 >> 1) + half];
        v2f bf; bf.x = bw.x; bf.y = bw.y;
        #pragma unroll
        for (int s = 0; s < SUB; ++s) {
            const float* ar = At + (s * 16 + ln) * AS + k0 + 2 * half;
            v2f a; a.x = ar[0]; a.y = ar[1];
            acc[s] = WMMA_F32(a, bf, acc[s]);
        }
    }

    // ---- epilogue: bias (+gelu) (+residual), guarded store ----
    const float bv = bias[n];
    #pragma unroll
    for (int s = 0; s < SUB; ++s) {
        #pragma unroll
        for (int r = 0; r < 8; ++r) {
            const int row = m0 + s * 16 + r + 8 * half;
            if (row < ROWS) {
                float v = acc[s][r] + bv;
                if (GELU) v = 0.5f * v * (1.f + erff(v * 0.70710678118654752f));
                if (RESID) v += resid[(size_t)row * nstride + n];
                out[(size_t)row * nstride + n] = v;
            }
        }
    }
}

// ---------------------------------------------------------------------------
// Flash attention over the factorized mask: band(|q-k|<=64) | q==0 | k==0.
// One wave per (q-tile, head, batch). qkv layout: [B*S, 768] (q | k | v).
// ---------------------------------------------------------------------------
__global__ void __launch_bounds__(32)
k_attn(const float* __restrict__ qkv, float* __restrict__ ctx)
{
    __shared__ float P[16 * 20];   // 16x16 probs, padded stride 20

    const int lane = threadIdx.x & 31;
    const int half = lane >> 4;
    const int ln   = lane & 15;
    const int qt   = blockIdx.x;
    const int h    = blockIdx.y;
    const int b    = blockIdx.z;
    const int q0   = qt * 16;
    const float scale = 0.125f;    // 1/sqrt(64)

    // preload all 16 Q A-fragments (whole dh=64) into registers, pre-scaled
    v2f qf[16];
    {
        const int qrow = min(q0 + ln, SEQ - 1);
        const float2* qp = (const float2*)(qkv + ((size_t)(b * SEQ + qrow)) * QKV + h * DH);
        #pragma unroll
        for (int i = 0; i < 16; ++i) {
            const float2 t = qp[i * 2 + half];
            qf[i].x = t.x * scale;
            qf[i].y = t.y * scale;
        }
    }

    v8f o0, o1, o2, o3;
    #pragma unroll
    for (int r = 0; r < 8; ++r) { o0[r] = 0.f; o1[r] = 0.f; o2[r] = 0.f; o3[r] = 0.f; }
    float mrow[8], lrow[8];
    #pragma unroll
    for (int r = 0; r < 8; ++r) { mrow[r] = -__builtin_huge_valf(); lrow[r] = 0.f; }

    int ktlo, kthi;
    if (q0 == 0) { ktlo = 0; kthi = QT - 1; }          // tile holds global query row
    else {
        ktlo = max(0, (q0 - WIN) >> 4);
        kthi = min(QT - 1, (q0 + 15 + WIN) >> 4);
    }

    auto process = [&](int kt) {
        // ---- S = Q @ K^T for this 16-key tile ----
        const int key  = kt * 16 + ln;
        const int keyc = min(key, SEQ - 1);
        const float2* kp = (const float2*)(qkv + ((size_t)(b * SEQ + keyc)) * QKV + DIM + h * DH);
        v8f s;
        #pragma unroll
        for (int r = 0; r < 8; ++r) s[r] = 0.f;
        #pragma unroll
        for (int i = 0; i < 16; ++i) {
            const float2 t = kp[i * 2 + half];
            v2f bf; bf.x = t.x; bf.y = t.y;
            s = WMMA_F32(qf[i], bf, s);
        }

        // ---- mask + online softmax (row reductions across 16-lane halves) ----
        #pragma unroll
        for (int r = 0; r < 8; ++r) {
            const int q = q0 + r + 8 * half;
            const int ad = (q > key) ? (q - key) : (key - q);
            const bool ok = (key < SEQ) && (q < SEQ) &&
                            ((q == 0) || (key == 0) || (ad <= WIN));
            float sv = ok ? s[r] : -__builtin_huge_valf();
            float mx = sv;
            mx = fmaxf(mx, __shfl_xor(mx, 1, 32));
            mx = fmaxf(mx, __shfl_xor(mx, 2, 32));
            mx = fmaxf(mx, __shfl_xor(mx, 4, 32));
            mx = fmaxf(mx, __shfl_xor(mx, 8, 32));
            const float mnew  = fmaxf(mrow[r], mx);
            float p = ok ? __expf(sv - mnew) : 0.f;
            const float alpha = __expf(mrow[r] - mnew);
            float ps = p;
            ps += __shfl_xor(ps, 1, 32);
            ps += __shfl_xor(ps, 2, 32);
            ps += __shfl_xor(ps, 4, 32);
            ps += __shfl_xor(ps, 8, 32);
            lrow[r] = lrow[r] * alpha + ps;
            mrow[r] = mnew;
            o0[r] *= alpha; o1[r] *= alpha; o2[r] *= alpha; o3[r] *= alpha;
            P[(r + 8 * half) * 20 + ln] = p;    // C-layout -> LDS for A-layout reads
        }
        __syncthreads();

        // ---- O += P @ V ----
        #pragma unroll
        for (int k0 = 0; k0 < 16; k0 += 4) {
            v2f a;
            a.x = P[ln * 20 + k0 + 2 * half];
            a.y = P[ln * 20 + k0 + 2 * half + 1];
            const int kr0 = min(kt * 16 + k0 + 2 * half, SEQ - 1);
            const int kr1 = min(kr0 + 1, SEQ - 1);
            const float* v0 = qkv + ((size_t)(b * SEQ + kr0)) * QKV + 2 * DIM + h * DH;
            const float* v1 = qkv + ((size_t)(b * SEQ + kr1)) * QKV + 2 * DIM + h * DH;
            { v2f bf; bf.x = v0[ln];      bf.y = v1[ln];      o0 = WMMA_F32(a, bf, o0); }
            { v2f bf; bf.x = v0[16 + ln]; bf.y = v1[16 + ln]; o1 = WMMA_F32(a, bf, o1); }
            { v2f bf; bf.x = v0[32 + ln]; bf.y = v1[32 + ln]; o2 = WMMA_F32(a, bf, o2); }
            { v2f bf; bf.x = v0[48 + ln]; bf.y = v1[48 + ln]; o3 = WMMA_F32(a, bf, o3); }
        }
        __syncthreads();
    };

    if (ktlo > 0) process(0);                       // global key column tile
    for (int kt = ktlo; kt <= kthi; ++kt) process(kt);

    // ---- normalize + store ctx ----
    #pragma unroll
    for (int r = 0; r < 8; ++r) {
        const int q = q0 + r + 8 * half;
        if (q < SEQ) {
            const float inv = 1.f / lrow[r];
            float* op = ctx + ((size_t)(b * SEQ + q)) * DIM + h * DH + ln;
            op[0]  = o0[r] * inv;
            op[16] = o1[r] * inv;
            op[32] = o2[r] * inv;
            op[48] = o3[r] * inv;
        }
    }
}

// ---------------------------------------------------------------------------
extern "C" void kernel_launch(void* const* d_in, const int* in_sizes, int n_in,
                              void* d_out, int out_size, void* d_ws, size_t ws_size,
                              hipStream_t stream)
{
    const float* x    = (const float*)d_in[0];
    const float* n1w  = (const float*)d_in[1];
    const float* inW  = (const float*)d_in[2];
    const float* inB  = (const float*)d_in[3];
    const float* outW = (const float*)d_in[4];
    const float* outB = (const float*)d_in[5];
    const float* n2w  = (const float*)d_in[6];
    const float* w1   = (const float*)d_in[7];
    const float* b1   = (const float*)d_in[8];
    const float* w2   = (const float*)d_in[9];
    const float* b2   = (const float*)d_in[10];
    float* out = (float*)d_out;

    float* qkv = (float*)d_ws;                    // ROWS * 768
    float* ctx = qkv + (size_t)ROWS * QKV;        // ROWS * 256
    float* x1  = ctx + (size_t)ROWS * DIM;        // ROWS * 256
    float* act = x1  + (size_t)ROWS * DIM;        // ROWS * 512

    const dim3 blk(256);
    const int MT32 = (ROWS + 31) / 32;  // 257
    const int MT16 = (ROWS + 15) / 16;  // 513

    // 1) qkv = rmsnorm(x, n1w) @ inW^T + inB
    k_gemm<DIM, 32, true, false, false><<<dim3(MT32, QKV / 128), blk, 0, stream>>>(
        x, n1w, inW, inB, nullptr, qkv, QKV);

    // 2) ctx = factorized attention(qkv)
    k_attn<<<dim3(QT, NH, BATCH), dim3(32), 0, stream>>>(qkv, ctx);

    // 3) x1 = x + ctx @ outW^T + outB
    k_gemm<DIM, 32, false, false, true><<<dim3(MT32, DIM / 128), blk, 0, stream>>>(
        ctx, nullptr, outW, outB, x, x1, DIM);

    // 4) act = gelu(rmsnorm(x1, n2w) @ w1^T + b1)
    k_gemm<DIM, 32, true, true, false><<<dim3(MT32, FF / 128), blk, 0, stream>>>(
        x1, n2w, w1, b1, nullptr, act, FF);

    // 5) out = x1 + act @ w2^T + b2
    k_gemm<FF, 16, false, false, true><<<dim3(MT16, DIM / 128), blk, 0, stream>>>(
        act, nullptr, w2, b2, x1, out, DIM);
}